// LUT1D_16226386444451
// MI455X (gfx1250) — compile-verified
//
#include <hip/hip_runtime.h>

// ---------------------------------------------------------------------------
// LUT1D forward: out[n,b] = (b0*w[b,i0] + b1*w[b,i0+1]) * x[n,b]   (complex)
// Pure streaming kernel: ~268 MB traffic -> ~11.5 us floor @ 23.3 TB/s.
// CDNA5 paths: global_load_async_to_lds_b64 + s_wait_asynccnt to stage the
// 2 KB weight LUT in LDS; NT-hinted b128 streaming loads/stores; 4 rows per
// thread (unrolled) to amortize LUT setup and deepen MLP against HBM latency.
// ---------------------------------------------------------------------------

typedef float v4f __attribute__((ext_vector_type(4)));
typedef float v2f __attribute__((ext_vector_type(2)));

// Types matching the async-to-LDS builtin signature:
//   void __builtin_amdgcn_global_load_async_to_lds_b64(v2i AS1*, v2i AS3*, Ii, Ii)
typedef int v2i __attribute__((vector_size(2 * sizeof(int))));
typedef __attribute__((address_space(1))) v2i gv2i_t;
typedef __attribute__((address_space(3))) v2i lv2i_t;

#define NBR 4
#define NSPL 64
#define LUT_STRIDE (NSPL + 1)   // +1 guard entry (zero) for the i0==63 tap pair
#define THREADS 256
#define ROWS_PER_THREAD 4
#define ROWS_PER_BLOCK (THREADS * ROWS_PER_THREAD)

__global__ __launch_bounds__(THREADS) void lut1d_fwd_kernel(
    const float* __restrict__ x_re, const float* __restrict__ x_im,
    const float* __restrict__ w_re, const float* __restrict__ w_im,
    float* __restrict__ out, int nrows)
{
    __shared__ float rawRe[NBR * NSPL];            // 1 KB
    __shared__ float rawIm[NBR * NSPL];            // 1 KB
    __shared__ v2f   lut[NBR * LUT_STRIDE];        // 2080 B interleaved (re,im)

    const int tid = threadIdx.x;

    // ---- Stage 1: async-copy both weight tables into LDS (CDNA5 path). ----
    // 256 threads x 8 bytes = 2 KB: tid<128 moves w_re, tid>=128 moves w_im.
    {
        const float* gsrc = (tid < 128) ? (w_re + 2 * tid)
                                        : (w_im + 2 * (tid - 128));
        float* ldst = (tid < 128) ? (rawRe + 2 * tid)
                                  : (rawIm + 2 * (tid - 128));
#if __has_builtin(__builtin_amdgcn_global_load_async_to_lds_b64)
        __builtin_amdgcn_global_load_async_to_lds_b64(
            (gv2i_t*)gsrc, (lv2i_t*)ldst, 0, 0);
#else
        unsigned lds_off = (unsigned)(unsigned long long)(lv2i_t*)ldst;
        asm volatile("global_load_async_to_lds_b64 %0, %1, off"
                     :: "v"(lds_off), "v"(gsrc) : "memory");
#endif
#if __has_builtin(__builtin_amdgcn_s_wait_asynccnt)
        __builtin_amdgcn_s_wait_asynccnt(0);
#else
        asm volatile("s_wait_asynccnt 0" ::: "memory");
#endif
    }
    __syncthreads();

    // ---- Stage 2: interleave into padded (re,im) LUT; zero the guard. ----
    // 260 entries: threads 0..255 do idx=tid, threads 0..3 also do 256+tid.
    {
        int idx = tid;
        {
            const int b = idx / LUT_STRIDE;
            const int k = idx - b * LUT_STRIDE;
            v2f v;
            v[0] = (k < NSPL) ? rawRe[b * NSPL + k] : 0.0f;
            v[1] = (k < NSPL) ? rawIm[b * NSPL + k] : 0.0f;
            lut[idx] = v;
        }
        idx = THREADS + tid;
        if (idx < NBR * LUT_STRIDE) {
            const int b = idx / LUT_STRIDE;
            const int k = idx - b * LUT_STRIDE;
            v2f v;
            v[0] = (k < NSPL) ? rawRe[b * NSPL + k] : 0.0f;
            v[1] = (k < NSPL) ? rawIm[b * NSPL + k] : 0.0f;
            lut[idx] = v;
        }
    }
    __syncthreads();

    // ---- Stage 3: 4 rows per thread, coalesced b128 NT streaming. ----
    const long long base = (long long)blockIdx.x * ROWS_PER_BLOCK + tid;
#pragma unroll
    for (int it = 0; it < ROWS_PER_THREAD; ++it) {
        const long long row = base + (long long)it * THREADS;
        if (row < nrows) {
            const v4f xr = __builtin_nontemporal_load((const v4f*)x_re + row);
            const v4f xi = __builtin_nontemporal_load((const v4f*)x_im + row);

            float ore[NBR], oim[NBR];
#pragma unroll
            for (int b = 0; b < NBR; ++b) {
                const float re = xr[b];
                const float im = xi[b];
                const float mag = sqrtf(fmaf(re, re, im * im));
                const float xa  = mag * (float)(NSPL - 1);
                const float k0  = floorf(xa);
                const float fr  = xa - k0;              // in [0,1)
                const int   i0  = (int)k0;              // >= 0 always
                // hat weights with the reference's out-of-range zeroing:
                //   b0 = relu(1-|xa-i0|)   zero when i0 > 63
                //   b1 = relu(1-|xa-i1|)   zero when i1 > 63  (i.e. i0 >= 63)
                const float b0 = (i0 <= NSPL - 1) ? (1.0f - fr) : 0.0f;
                const float b1 = (i0 <= NSPL - 2) ? fr : 0.0f;
                const int i0c = (i0 < NSPL - 1) ? i0 : (NSPL - 1);
                // adjacent b64 LDS loads -> ds_load_2addr_b64 candidate
                const v2f w0 = lut[b * LUT_STRIDE + i0c];
                const v2f w1 = lut[b * LUT_STRIDE + i0c + 1];
                const float zre = fmaf(b0, w0[0], b1 * w1[0]);
                const float zim = fmaf(b0, w0[1], b1 * w1[1]);
                // out = z * x (complex)
                ore[b] = fmaf(zre, re, -(zim * im));
                oim[b] = fmaf(zre, im,   zim * re);
            }

            v4f oA, oB;
            oA[0] = ore[0]; oA[1] = oim[0]; oA[2] = ore[1]; oA[3] = oim[1];
            oB[0] = ore[2]; oB[1] = oim[2]; oB[2] = ore[3]; oB[3] = oim[3];
            __builtin_nontemporal_store(oA, (v4f*)out + 2 * row);
            __builtin_nontemporal_store(oB, (v4f*)out + 2 * row + 1);
        }
    }
}

extern "C" void kernel_launch(void* const* d_in, const int* in_sizes, int n_in,
                              void* d_out, int out_size, void* d_ws, size_t ws_size,
                              hipStream_t stream) {
    (void)n_in; (void)out_size; (void)d_ws; (void)ws_size;
    const float* x_re = (const float*)d_in[0];   // [N, 4]
    const float* x_im = (const float*)d_in[1];   // [N, 4]
    const float* w_re = (const float*)d_in[2];   // [4, 64]
    const float* w_im = (const float*)d_in[3];   // [4, 64]
    float* out = (float*)d_out;                  // [N, 4] complex64 -> interleaved f32

    const int nrows = in_sizes[0] / NBR;         // N
    const int blocks = (nrows + ROWS_PER_BLOCK - 1) / ROWS_PER_BLOCK;
    hipLaunchKernelGGL(lut1d_fwd_kernel, dim3(blocks), dim3(THREADS), 0, stream,
                       x_re, x_im, w_re, w_im, out, nrows);
}